// GraphTransformer_13460427506264
// MI455X (gfx1250) — compile-verified
//
#include <hip/hip_runtime.h>
#include <math.h>

// ---------------------------------------------------------------------------
// Graph transformer for MI455X (gfx1250, wave32).
// All GEMMs use V_WMMA_F32_16X16X4_F32 (f32 data -> f32 WMMA path, no
// precision loss). One wave owns one 16x16 output tile; K advances 4/step.
// ---------------------------------------------------------------------------

#define BB 4
#define NN 512
#define HH 64
#define NHH 4
#define HDM 16
#define LLAYERS 3
#define NODEF 18
#define EDGEF 7
#define H4 16
#define FFD 256
#define NEGV -1.0e9f

typedef __attribute__((ext_vector_type(2))) float v2f;
typedef __attribute__((ext_vector_type(8))) float v8f;

// ---------------------------------------------------------------------------
// Generic row-major GEMM: C[M,N] = act(A[M,K] @ B[K,N] + bias[N])
// act: 0 = none, 1 = exact GELU.
// WMMA fragments (per CDNA5 ISA):
//   A 16x4 f32: lanes 0-15 -> M=lane, VGPR0=K0, VGPR1=K1; lanes 16-31 -> K2,K3
//   C/D: VGPR r, lanes 0-15 -> M=r; lanes 16-31 -> M=r+8; N = lane&15
// Partial M tiles: clamp A-row loads, guard stores (EXEC stays all-ones).
// ---------------------------------------------------------------------------
__global__ void gemm_wmma(const float* __restrict__ A, int lda,
                          const float* __restrict__ Bm, int ldb,
                          const float* __restrict__ bias,
                          float* __restrict__ C, int ldc,
                          int M, int K, int N, int act)
{
    const int lane = threadIdx.x & 31;
    const int wave = threadIdx.x >> 5;
    const int tilesN = N >> 4;
    const int tilesM = (M + 15) >> 4;
    const long tile = (long)blockIdx.x * 8 + wave;
    if (tile >= (long)tilesM * tilesN) return;
    const int tM = (int)(tile / tilesN);
    const int tN = (int)(tile % tilesN);
    const int half = lane >> 4;
    const int l16  = lane & 15;

    int rowA = tM * 16 + l16;
    if (rowA >= M) rowA = M - 1;                 // clamp; results discarded
    const int col = tN * 16 + l16;

    const float* ap = A  + (long)rowA * lda + 2 * half;
    const float* bp = Bm + (long)(2 * half) * ldb + col;

    v8f acc = {};
    for (int k = 0; k < K; k += 4) {
        v2f a, b;
        a.x = ap[k];
        a.y = ap[k + 1];
        b.x = bp[(long)k * ldb];
        b.y = bp[(long)(k + 1) * ldb];
        acc = __builtin_amdgcn_wmma_f32_16x16x4_f32(
            false, a, false, b, (short)0, acc, false, false);
    }

    const float bv = bias ? bias[col] : 0.0f;
    for (int r = 0; r < 8; ++r) {
        const int rowC = tM * 16 + r + 8 * half;
        if (rowC < M) {
            float v = acc[r] + bv;
            if (act == 1) v = 0.5f * v * (1.0f + erff(v * 0.70710678118654752f));
            C[(long)rowC * ldc + col] = v;
        }
    }
}

// ---------------------------------------------------------------------------
// scores[bh,i,j] = nmask[b,j] ? 0.25 * sum_d Q[b,i,h,d] K[b,j,h,d] + bias : NEG
// grid (4, 32, 16): x*8waves -> tileJ, y -> tileI, z -> b*NH+h
// ---------------------------------------------------------------------------
__global__ void scores_kernel(const float* __restrict__ Q,
                              const float* __restrict__ Kt,
                              const float* __restrict__ bias,
                              const unsigned char* __restrict__ nmask,
                              float* __restrict__ S)
{
    const int lane = threadIdx.x & 31;
    const int wave = threadIdx.x >> 5;
    const int tJ = blockIdx.x * 8 + wave;
    const int tI = blockIdx.y;
    const int bh = blockIdx.z;
    const int b = bh >> 2, hh = bh & 3;
    const int half = lane >> 4, l16 = lane & 15;

    const int i0 = tI * 16 + l16;
    const int j  = tJ * 16 + l16;
    const float* qp = Q  + ((long)(b * NN + i0) * NHH + hh) * HDM + 2 * half;
    const float* kp = Kt + ((long)(b * NN + j ) * NHH + hh) * HDM + 2 * half;

    v8f acc = {};
    for (int k = 0; k < HDM; k += 4) {
        v2f a, bv;
        a.x  = qp[k];  a.y  = qp[k + 1];
        bv.x = kp[k];  bv.y = kp[k + 1];
        acc = __builtin_amdgcn_wmma_f32_16x16x4_f32(
            false, a, false, bv, (short)0, acc, false, false);
    }

    const bool mj = nmask[b * NN + j] != 0;
    for (int r = 0; r < 8; ++r) {
        const int i = tI * 16 + r + 8 * half;
        const long idx = ((long)bh * NN + i) * NN + j;
        const float v = acc[r] * 0.25f + bias[idx];
        S[idx] = mj ? v : NEGV;
    }
}

// ---------------------------------------------------------------------------
// O[b,i,h*16+d] = sum_j P[bh,i,j] * V[b,j,h,d]     (K=512)
// grid (4, 16): x*8waves -> tileI, y -> bh
// ---------------------------------------------------------------------------
__global__ void attnv_kernel(const float* __restrict__ P,
                             const float* __restrict__ V,
                             float* __restrict__ O)
{
    const int lane = threadIdx.x & 31;
    const int wave = threadIdx.x >> 5;
    const int tI = blockIdx.x * 8 + wave;
    const int bh = blockIdx.y;
    const int b = bh >> 2, hh = bh & 3;
    const int half = lane >> 4, l16 = lane & 15;

    const int i0 = tI * 16 + l16;
    const int d  = l16;
    const float* ap = P + ((long)bh * NN + i0) * NN + 2 * half;

    v8f acc = {};
    for (int k = 0; k < NN; k += 4) {
        const int j0 = k + 2 * half;
        v2f a, bv;
        a.x  = ap[k];  a.y = ap[k + 1];
        bv.x = V[((long)(b * NN + j0    ) * NHH + hh) * HDM + d];
        bv.y = V[((long)(b * NN + j0 + 1) * NHH + hh) * HDM + d];
        acc = __builtin_amdgcn_wmma_f32_16x16x4_f32(
            false, a, false, bv, (short)0, acc, false, false);
    }
    for (int r = 0; r < 8; ++r) {
        const int i = tI * 16 + r + 8 * half;
        O[(long)(b * NN + i) * HH + hh * HDM + d] = acc[r];
    }
}

// ------------------------- elementwise / small kernels ---------------------

__global__ void node_proj_kernel(const float* __restrict__ nf,
                                 const float* __restrict__ W,
                                 const float* __restrict__ bvec,
                                 float* __restrict__ h)
{
    const int row = blockIdx.x;          // B*N rows
    const int o = threadIdx.x;           // 64
    float s = bvec[o];
    const float* x = nf + (long)row * NODEF;
    for (int f = 0; f < NODEF; ++f) s += x[f] * W[f * HH + o];
    h[(long)row * HH + o] = s;
}

// Fold edge_proj into eb1: Wc[l](7x16), bc[l](16). One block.
__global__ void combine_kernel(const float* __restrict__ epw,  // (7,64)
                               const float* __restrict__ epb,  // (64)
                               const float* __restrict__ eb1w, // (L,64,16)
                               const float* __restrict__ eb1b, // (L,16)
                               float* __restrict__ Wc,         // (L,7,16)
                               float* __restrict__ bc)         // (L,16)
{
    const int tid = threadIdx.x;
    if (tid < LLAYERS * EDGEF * H4) {
        const int l = tid / (EDGEF * H4);
        const int r = tid % (EDGEF * H4);
        const int f = r / H4, o = r % H4;
        float s = 0.0f;
        for (int c = 0; c < HH; ++c)
            s += epw[f * HH + c] * eb1w[((long)l * HH + c) * H4 + o];
        Wc[tid] = s;
    } else if (tid < LLAYERS * EDGEF * H4 + LLAYERS * H4) {
        const int idx = tid - LLAYERS * EDGEF * H4;
        const int l = idx / H4, o = idx % H4;
        float s = eb1b[l * H4 + o];
        for (int c = 0; c < HH; ++c)
            s += epb[c] * eb1w[((long)l * HH + c) * H4 + o];
        bc[idx] = s;
    }
}

__global__ void spd_init_kernel(const unsigned char* __restrict__ emask,
                                float* __restrict__ s)
{
    const long idx = (long)blockIdx.x * blockDim.x + threadIdx.x;
    const int ij = (int)(idx % ((long)NN * NN));
    const int i = ij / NN, j = ij % NN;
    s[idx] = (i == j) ? 0.0f : (emask[idx] ? 1.0f : (float)NN);
}

// One FW relaxation step for pivot k. blockIdx.x = b*N+i, threadIdx.x = j.
__global__ void fw_step_kernel(float* __restrict__ s, int k)
{
    const int bi = blockIdx.x;
    const int b = bi / NN, i = bi % NN;
    const int j = threadIdx.x;
    const float sik = s[((long)b * NN + i) * NN + k];
    const float skj = s[((long)b * NN + k) * NN + j];
    float* p = &s[((long)b * NN + i) * NN + j];
    const float alt = sik + skj;
    if (alt < *p) *p = alt;
}

__device__ __forceinline__ float leaky_f(float x) {
    return x >= 0.0f ? x : 0.2f * x;
}

// Per-edge bias for one layer: edge-MLP (LN over 16 in registers) vs SPD-MLP,
// selected by edge_mask. Writes bias[(b*NH+h)*N*N + i*N + j] for h=0..3.
__global__ void bias_kernel(const float* __restrict__ ef,
                            const unsigned char* __restrict__ emask,
                            const float* __restrict__ spd,
                            const float* __restrict__ Wc,   // (7,16) layer slice
                            const float* __restrict__ bc,   // (16)
                            const float* __restrict__ lng,
                            const float* __restrict__ lnb,
                            const float* __restrict__ w2,   // (16,4)
                            const float* __restrict__ b2,   // (4)
                            const float* __restrict__ sw1,  // (16)
                            const float* __restrict__ sb1,  // (16)
                            const float* __restrict__ sw2,  // (16,4)
                            const float* __restrict__ sb2,  // (4)
                            float* __restrict__ bias)
{
    const long idx = (long)blockIdx.x * blockDim.x + threadIdx.x;   // B*N*N
    const int b  = (int)(idx / ((long)NN * NN));
    const int ij = (int)(idx % ((long)NN * NN));
    const int i = ij / NN, j = ij % NN;

    // edge branch: x = leaky(LN(ef@Wc + bc))
    float x[H4];
    for (int o = 0; o < H4; ++o) x[o] = bc[o];
    const float* e = ef + idx * EDGEF;
    for (int f = 0; f < EDGEF; ++f) {
        const float ev = e[f];
        for (int o = 0; o < H4; ++o) x[o] += ev * Wc[f * H4 + o];
    }
    float m = 0.0f;
    for (int o = 0; o < H4; ++o) m += x[o];
    m *= (1.0f / H4);
    float vv = 0.0f;
    for (int o = 0; o < H4; ++o) { const float d = x[o] - m; vv += d * d; }
    vv *= (1.0f / H4);
    const float rs = rsqrtf(vv + 1e-5f);
    float eb[NHH];
    for (int hh = 0; hh < NHH; ++hh) eb[hh] = b2[hh];
    for (int o = 0; o < H4; ++o) {
        const float y = leaky_f((x[o] - m) * rs * lng[o] + lnb[o]);
        for (int hh = 0; hh < NHH; ++hh) eb[hh] += y * w2[o * NHH + hh];
    }

    // spd branch
    const float sn = fminf(spd[idx], 10.0f) * 0.1f;   // spd >= 0 always
    float sb[NHH];
    for (int hh = 0; hh < NHH; ++hh) sb[hh] = sb2[hh];
    for (int o = 0; o < H4; ++o) {
        const float z = leaky_f(sn * sw1[o] + sb1[o]);
        for (int hh = 0; hh < NHH; ++hh) sb[hh] += z * sw2[o * NHH + hh];
    }

    const bool em = emask[idx] != 0;
    for (int hh = 0; hh < NHH; ++hh)
        bias[((long)(b * NHH + hh) * NN + i) * NN + j] = em ? eb[hh] : sb[hh];
}

// Row softmax over N=512, one block (256 threads) per (bh,i) row, in place.
__global__ void softmax_rows_kernel(float* __restrict__ S)
{
    __shared__ float red[256];
    float* p = S + (long)blockIdx.x * NN;
    const int t = threadIdx.x;
    const float a0 = p[t], a1 = p[t + 256];
    red[t] = fmaxf(a0, a1);
    __syncthreads();
    for (int s = 128; s > 0; s >>= 1) {
        if (t < s) red[t] = fmaxf(red[t], red[t + s]);
        __syncthreads();
    }
    const float mx = red[0];
    __syncthreads();
    const float e0 = expf(a0 - mx), e1 = expf(a1 - mx);
    red[t] = e0 + e1;
    __syncthreads();
    for (int s = 128; s > 0; s >>= 1) {
        if (t < s) red[t] += red[t + s];
        __syncthreads();
    }
    const float inv = 1.0f / red[0];
    p[t] = e0 * inv;
    p[t + 256] = e1 * inv;
}

// out = LN(x [+ res]) * g + b, width=64, optional leaky. blockDim = 64.
__global__ void ln_kernel(const float* __restrict__ x,
                          const float* __restrict__ res,
                          const float* __restrict__ g,
                          const float* __restrict__ bta,
                          float* __restrict__ out, int leaky)
{
    __shared__ float red[HH];
    const long row = blockIdx.x;
    const int t = threadIdx.x;
    float v = x[row * HH + t];
    if (res) v += res[row * HH + t];
    red[t] = v;
    __syncthreads();
    for (int s = HH >> 1; s > 0; s >>= 1) {
        if (t < s) red[t] += red[t + s];
        __syncthreads();
    }
    const float mean = red[0] * (1.0f / HH);
    __syncthreads();
    const float d = v - mean;
    red[t] = d * d;
    __syncthreads();
    for (int s = HH >> 1; s > 0; s >>= 1) {
        if (t < s) red[t] += red[t + s];
        __syncthreads();
    }
    const float var = red[0] * (1.0f / HH);
    float y = d * rsqrtf(var + 1e-5f) * g[t] + bta[t];
    if (leaky) y = leaky_f(y);
    out[row * HH + t] = y;
}

// hc[b,n] = concat(h[b,0], h[b,1+n]); flat over B*(N-1)*2H
__global__ void build_hc_kernel(const float* __restrict__ h,
                                float* __restrict__ hc)
{
    const long idx = (long)blockIdx.x * blockDim.x + threadIdx.x;
    if (idx >= (long)BB * (NN - 1) * 2 * HH) return;
    const int row = (int)(idx / (2 * HH));
    const int c = (int)(idx % (2 * HH));
    const int b = row / (NN - 1);
    const int n = row % (NN - 1);
    const float v = (c < HH) ? h[((long)b * NN) * HH + c]
                             : h[((long)b * NN + 1 + n) * HH + (c - HH)];
    hc[idx] = v;
}

// logits[row] = x[row,:] . w + b ; blockDim = 64
__global__ void logits_kernel(const float* __restrict__ x,
                              const float* __restrict__ w,
                              const float* __restrict__ b2,
                              float* __restrict__ lg)
{
    __shared__ float red[HH];
    const long row = blockIdx.x;
    const int t = threadIdx.x;
    red[t] = x[row * HH + t] * w[t];
    __syncthreads();
    for (int s = HH >> 1; s > 0; s >>= 1) {
        if (t < s) red[t] += red[t + s];
        __syncthreads();
    }
    if (t == 0) lg[row] = red[0] + b2[0];
}

// Candidate masking + softmax over 511 logits per batch. blockDim = 512.
__global__ void final_softmax_kernel(const float* __restrict__ lg,
                                     const unsigned char* __restrict__ nmask,
                                     float* __restrict__ probs)
{
    __shared__ float red[512];
    const int b = blockIdx.x;
    const int t = threadIdx.x;
    const bool active = t < (NN - 1);
    int cand = 0;
    if (active) cand = nmask[b * NN + 1 + t] ? 1 : 0;
    red[t] = (float)cand;
    __syncthreads();
    for (int s = 256; s > 0; s >>= 1) {
        if (t < s) red[t] += red[t + s];
        __syncthreads();
    }
    const bool nocand = red[0] == 0.0f;
    __syncthreads();
    if (nocand && t == 0) cand = 1;

    float val = NEGV;
    if (active && cand) val = lg[b * (NN - 1) + t];
    red[t] = active ? val : -3.0e38f;
    __syncthreads();
    for (int s = 256; s > 0; s >>= 1) {
        if (t < s) red[t] = fmaxf(red[t], red[t + s]);
        __syncthreads();
    }
    const float mx = red[0];
    __syncthreads();
    const float e = active ? expf(val - mx) : 0.0f;
    red[t] = e;
    __syncthreads();
    for (int s = 256; s > 0; s >>= 1) {
        if (t < s) red[t] += red[t + s];
        __syncthreads();
    }
    if (active) probs[b * (NN - 1) + t] = e / red[0];
}

__global__ void copy_kernel(const float* __restrict__ src,
                            float* __restrict__ dst, long n)
{
    const long idx = (long)blockIdx.x * blockDim.x + threadIdx.x;
    if (idx < n) dst[idx] = src[idx];
}

// ---------------------------------------------------------------------------

extern "C" void kernel_launch(void* const* d_in, const int* in_sizes, int n_in,
                              void* d_out, int out_size, void* d_ws, size_t ws_size,
                              hipStream_t stream)
{
    const float* node_features = (const float*)d_in[0];
    const float* edge_features = (const float*)d_in[1];
    const unsigned char* node_mask = (const unsigned char*)d_in[2];
    const unsigned char* edge_mask = (const unsigned char*)d_in[3];
    const float* node_proj_w = (const float*)d_in[4];
    const float* node_proj_b = (const float*)d_in[5];
    const float* edge_proj_w = (const float*)d_in[6];
    const float* edge_proj_b = (const float*)d_in[7];
    const float* wq   = (const float*)d_in[8];
    const float* wk   = (const float*)d_in[9];
    const float* wv   = (const float*)d_in[10];
    const float* eb1_w = (const float*)d_in[11];
    const float* eb1_b = (const float*)d_in[12];
    const float* eb_ln_g = (const float*)d_in[13];
    const float* eb_ln_b = (const float*)d_in[14];
    const float* eb2_w = (const float*)d_in[15];
    const float* eb2_b = (const float*)d_in[16];
    const float* spd1_w = (const float*)d_in[17];
    const float* spd1_b = (const float*)d_in[18];
    const float* spd2_w = (const float*)d_in[19];
    const float* spd2_b = (const float*)d_in[20];
    const float* wo_w = (const float*)d_in[21];
    const float* wo_b = (const float*)d_in[22];
    const float* ln1_g = (const float*)d_in[23];
    const float* ln1_b = (const float*)d_in[24];
    const float* ln2_g = (const float*)d_in[25];
    const float* ln2_b = (const float*)d_in[26];
    const float* ffn1_w = (const float*)d_in[27];
    const float* ffn1_b = (const float*)d_in[28];
    const float* ffn2_w = (const float*)d_in[29];
    const float* ffn2_b = (const float*)d_in[30];
    const float* score1_w = (const float*)d_in[31];
    const float* score1_b = (const float*)d_in[32];
    const float* score_ln_g = (const float*)d_in[33];
    const float* score_ln_b = (const float*)d_in[34];
    const float* score2_w = (const float*)d_in[35];
    const float* score2_b = (const float*)d_in[36];

    float* ws = (float*)d_ws;
    size_t off = 0;
    float* spd  = ws + off; off += (size_t)BB * NN * NN;                // 4 MB
    float* bias = ws + off; off += (size_t)BB * NHH * NN * NN;          // 16.8 MB
    float* attn = ws + off; off += (size_t)BB * NHH * NN * NN;          // 16.8 MB
    float* h    = ws + off; off += (size_t)BB * NN * HH;
    float* q    = ws + off; off += (size_t)BB * NN * HH;
    float* kk   = ws + off; off += (size_t)BB * NN * HH;
    float* v    = ws + off; off += (size_t)BB * NN * HH;
    float* ao   = ws + off; off += (size_t)BB * NN * HH;
    float* po   = ws + off; off += (size_t)BB * NN * HH;
    float* ff1  = ws + off; off += (size_t)BB * NN * FFD;
    float* ff2  = ws + off; off += (size_t)BB * NN * HH;
    float* Wc   = ws + off; off += (size_t)LLAYERS * EDGEF * H4;
    float* bc   = ws + off; off += (size_t)LLAYERS * H4;
    float* hcb  = ws + off; off += (size_t)BB * (NN - 1) * 2 * HH;
    float* s1   = ws + off; off += (size_t)BB * (NN - 1) * HH;
    float* lg   = ws + off; off += 2048;

    const int Mrows = BB * NN;   // 2048
    auto gblocks = [](int M, int N) {
        const int t = ((M + 15) / 16) * (N / 16);
        return (t + 7) / 8;
    };

    // h = node_features @ W + b
    node_proj_kernel<<<Mrows, HH, 0, stream>>>(node_features, node_proj_w,
                                               node_proj_b, h);
    // fold edge_proj into eb1 per layer
    combine_kernel<<<1, 512, 0, stream>>>(edge_proj_w, edge_proj_b,
                                          eb1_w, eb1_b, Wc, bc);
    // Floyd-Warshall (4 MB working set, L2-resident)
    spd_init_kernel<<<(BB * NN * NN) / 256, 256, 0, stream>>>(edge_mask, spd);
    for (int kp = 0; kp < NN; ++kp)
        fw_step_kernel<<<BB * NN, NN, 0, stream>>>(spd, kp);

    for (int l = 0; l < LLAYERS; ++l) {
        bias_kernel<<<(BB * NN * NN) / 256, 256, 0, stream>>>(
            edge_features, edge_mask, spd,
            Wc + l * EDGEF * H4, bc + l * H4,
            eb_ln_g + l * H4, eb_ln_b + l * H4,
            eb2_w + l * H4 * NHH, eb2_b + l * NHH,
            spd1_w + l * H4, spd1_b + l * H4,
            spd2_w + l * H4 * NHH, spd2_b + l * NHH, bias);

        // Q, K, V projections (no bias in reference)
        gemm_wmma<<<gblocks(Mrows, HH), 256, 0, stream>>>(
            h, HH, wq + l * HH * HH, HH, nullptr, q, HH, Mrows, HH, HH, 0);
        gemm_wmma<<<gblocks(Mrows, HH), 256, 0, stream>>>(
            h, HH, wk + l * HH * HH, HH, nullptr, kk, HH, Mrows, HH, HH, 0);
        gemm_wmma<<<gblocks(Mrows, HH), 256, 0, stream>>>(
            h, HH, wv + l * HH * HH, HH, nullptr, v, HH, Mrows, HH, HH, 0);

        scores_kernel<<<dim3(4, NN / 16, BB * NHH), 256, 0, stream>>>(
            q, kk, bias, node_mask, attn);
        softmax_rows_kernel<<<BB * NHH * NN, 256, 0, stream>>>(attn);
        attnv_kernel<<<dim3(4, BB * NHH), 256, 0, stream>>>(attn, v, ao);

        gemm_wmma<<<gblocks(Mrows, HH), 256, 0, stream>>>(
            ao, HH, wo_w + l * HH * HH, HH, wo_b + l * HH, po, HH,
            Mrows, HH, HH, 0);
        ln_kernel<<<Mrows, HH, 0, stream>>>(po, h, ln1_g + l * HH,
                                            ln1_b + l * HH, h, 0);

        gemm_wmma<<<gblocks(Mrows, FFD), 256, 0, stream>>>(
            h, HH, ffn1_w + l * HH * FFD, FFD, ffn1_b + l * FFD, ff1, FFD,
            Mrows, HH, FFD, 1 /*gelu*/);
        gemm_wmma<<<gblocks(Mrows, HH), 256, 0, stream>>>(
            ff1, FFD, ffn2_w + l * FFD * HH, HH, ffn2_b + l * HH, ff2, HH,
            Mrows, FFD, HH, 0);
        ln_kernel<<<Mrows, HH, 0, stream>>>(ff2, h, ln2_g + l * HH,
                                            ln2_b + l * HH, h, 0);
    }

    // Scoring head
    const int Mhc = BB * (NN - 1);   // 2044 (partial tiles guarded)
    build_hc_kernel<<<((long)Mhc * 2 * HH + 255) / 256, 256, 0, stream>>>(h, hcb);
    gemm_wmma<<<gblocks(Mhc, HH), 256, 0, stream>>>(
        hcb, 2 * HH, score1_w, HH, score1_b, s1, HH, Mhc, 2 * HH, HH, 0);
    ln_kernel<<<Mhc, HH, 0, stream>>>(s1, nullptr, score_ln_g, score_ln_b,
                                      s1, 1 /*leaky*/);
    logits_kernel<<<Mhc, HH, 0, stream>>>(s1, score2_w, score2_b, lg);

    float* out = (float*)d_out;
    final_softmax_kernel<<<BB, 512, 0, stream>>>(lg, node_mask, out);
    copy_kernel<<<((long)BB * NN * HH + 255) / 256, 256, 0, stream>>>(
        h, out + (size_t)BB * (NN - 1), (long)BB * NN * HH);
}